// LegendreBasis_57329223467185
// MI455X (gfx1250) — compile-verified
//
#include <hip/hip_runtime.h>

// CDNA5 / gfx1250: wave32, V_WMMA_F32_16X16X4_F32.
// out[b,j] = prod_d P_{e[d,j]}(clip(actions[b,d]))
//
// Gather-as-one-hot-GEMM (exact in f32):
//   g_d = A_d(16x4: P1..P4) x B_d(4x16: onehot of e in 1..4) + 0
//   gather_d = g_d + 1[e_d==0]        (P0 == 1; g_d is exactly 0 when e_d==0)
//   out tile = prod_d gather_d        (balanced tree of packed f32 muls)
//
// Each wave owns 2 row-tiles (32 batch rows) and sweeps all 31 column tiles,
// so the one-hot B/c0 build amortizes over 16 WMMAs. Output is a 260 MB
// write-once stream -> nontemporal stores (true roofline limit ~11 us @23.3TB/s).

typedef float v2f __attribute__((ext_vector_type(2)));
typedef float v8f __attribute__((ext_vector_type(8)));

#define DIMS 8
#define LDS_STRIDE 512   // >= NB (495), padded; keeps tail reads in-bounds
#define RT 2             // 16-row tiles per wave

__global__ __launch_bounds__(256)
void legendre_basis_wmma_kernel(const float* __restrict__ actions,
                                const int*   __restrict__ exponents,
                                float*       __restrict__ out,
                                int batch, int nb)
{
    __shared__ int sExp[DIMS * LDS_STRIDE];

    // Stage exponent table [DIMS, nb] into LDS once per block.
    for (int i = threadIdx.x; i < DIMS * nb; i += blockDim.x) {
        int d = i / nb;
        int j = i - d * nb;
        sExp[d * LDS_STRIDE + j] = exponents[i];
    }
    __syncthreads();

    const int lane = threadIdx.x & 31;
    const int wid  = threadIdx.x >> 5;
    const int gw   = blockIdx.x * 8 + wid;     // global wave id
    const int rowBase = gw * (16 * RT);        // 32 batch rows per wave
    if (rowBase >= batch) return;              // wave-uniform (EXEC all-1 for WMMA)

    const bool hi  = lane >= 16;
    const int mrow = lane & 15;                // A-matrix row owned by this lane
    const int col  = lane & 15;                // B/D-matrix column owned by this lane

    // A operands: per row-tile, per dim, Legendre P1..P4 of the clipped action.
    v2f a[RT][DIMS];
    #pragma unroll
    for (int r = 0; r < RT; ++r) {
        const float4* arow =
            (const float4*)(actions + (size_t)(rowBase + r * 16 + mrow) * DIMS);
        float4 x0 = arow[0];
        float4 x1 = arow[1];
        float xs[DIMS] = {x0.x, x0.y, x0.z, x0.w, x1.x, x1.y, x1.z, x1.w};
        #pragma unroll
        for (int d = 0; d < DIMS; ++d) {
            float x  = fminf(fmaxf(xs[d], -1.0f), 1.0f);
            float p1 = x;                                            // P1
            float p2 = 1.5f * x * p1 - 0.5f;                         // P2
            float p3 = (5.0f / 3.0f) * x * p2 - (2.0f / 3.0f) * p1;  // P3
            float p4 = 1.75f * x * p3 - 0.75f * p2;                  // P4
            a[r][d] = hi ? (v2f){p3, p4} : (v2f){p1, p2};            // K=2,3 : K=0,1
        }
    }

    const v8f vzero = {0, 0, 0, 0, 0, 0, 0, 0};
    const int ntiles = (nb + 15) >> 4;         // 31 for nb=495

    for (int t = 0; t < ntiles; ++t) {
        const int j = t * 16 + col;            // j <= 495 < LDS_STRIDE: read safe

        // One-hot selection operands: built once, reused by RT row-tiles.
        v2f   b[DIMS];
        float c0[DIMS];
        #pragma unroll
        for (int d = 0; d < DIMS; ++d) {
            const int e = sExp[d * LDS_STRIDE + j];
            b[d]  = hi ? (v2f){e == 3 ? 1.0f : 0.0f, e == 4 ? 1.0f : 0.0f}
                       : (v2f){e == 1 ? 1.0f : 0.0f, e == 2 ? 1.0f : 0.0f};
            c0[d] = (e == 0) ? 1.0f : 0.0f;
        }

        #pragma unroll
        for (int r = 0; r < RT; ++r) {
            v8f g[DIMS];
            #pragma unroll
            for (int d = 0; d < DIMS; ++d) {
                g[d] = __builtin_amdgcn_wmma_f32_16x16x4_f32(
                    false, a[r][d], false, b[d], (short)0, vzero, false, false);
            }
            // gather_d = g_d + 1[e_d==0]; cross-dim product as balanced tree.
            v8f f0 = (g[0] + c0[0]) * (g[1] + c0[1]);
            v8f f1 = (g[2] + c0[2]) * (g[3] + c0[3]);
            v8f f2 = (g[4] + c0[4]) * (g[5] + c0[5]);
            v8f f3 = (g[6] + c0[6]) * (g[7] + c0[7]);
            v8f prod = (f0 * f1) * (f2 * f3);

            if (j < nb) {
                // D layout: VGPR k -> row (k + 8*hi), col = lane&15.
                float* op = out + (size_t)(rowBase + r * 16 + (hi ? 8 : 0)) * nb + j;
                #pragma unroll
                for (int k = 0; k < 8; ++k) {
                    __builtin_nontemporal_store(prod[k], op + (size_t)k * nb);
                }
            }
        }
    }
}

extern "C" void kernel_launch(void* const* d_in, const int* in_sizes, int n_in,
                              void* d_out, int out_size, void* d_ws, size_t ws_size,
                              hipStream_t stream)
{
    const float* actions   = (const float*)d_in[0];   // [B, 8] f32
    const int*   exponents = (const int*)  d_in[1];   // [8, NB] i32
    float*       out       = (float*)d_out;           // [B, NB] f32

    const int batch = in_sizes[0] / DIMS;             // 131072
    const int nb    = in_sizes[1] / DIMS;             // 495

    // 256 threads = 8 waves; each wave owns 32 rows -> 256 rows per block.
    const int blocks = (batch + (16 * RT * 8) - 1) / (16 * RT * 8);
    legendre_basis_wmma_kernel<<<blocks, 256, 0, stream>>>(actions, exponents,
                                                           out, batch, nb);
}